// DMTetGeometry_17042430230583
// MI455X (gfx1250) — compile-verified
//
#include <hip/hip_runtime.h>
#include <hip/hip_bf16.h>
#include <stdint.h>

// ---------------------------------------------------------------------------
// DMTet on MI455X (gfx1250). Memory/sort-bound: ~1.3 GB traffic -> ~60us floor
// at 23.3 TB/s. No matmul => WMMA is not applicable; the CDNA5 paths exercised
// are the async global->LDS data mover (global_load_async_to_lds_b128/b64 +
// s_wait_asynccnt, ASYNCcnt), wave32 ballot multisplit, global_prefetch_b8.
// Round-2 change: hierarchical (3-kernel) scan replaces the serial
// single-block scan of the 562k-entry digit histogram; async staging widened
// to b128.
// ---------------------------------------------------------------------------

#define NVERTS 300000
#define NTETS  1500000
#define KEDGE  (6 * NTETS)                 // 9,000,000 edge slots
#define GRID_RES 128

static constexpr unsigned long long SENT_KEY =
    (unsigned long long)NVERTS * (unsigned long long)NVERTS;  // 9.0e10, 37 bits

// ---- radix sort config: sort bits [24..63] of packed (key<<24 | slot) ------
#define RS_THREADS 256
#define RS_ITEMS   16
#define RS_TILE    (RS_THREADS * RS_ITEMS)                 // 4096
#define RS_BLOCKS  ((KEDGE + RS_TILE - 1) / RS_TILE)       // 2198
#define RS_DIGITS  256
#define RS_PASSES  5
#define RS_SHIFT0  24

#define SCAN_TILE  4096                                    // 256 thr x 16 items

// ---- marching-tets tables --------------------------------------------------
__device__ const int d_tri_table[16][6] = {
    {-1,-1,-1,-1,-1,-1}, { 1, 0, 2,-1,-1,-1}, { 4, 0, 3,-1,-1,-1},
    { 1, 4, 2, 1, 3, 4}, { 3, 1, 5,-1,-1,-1}, { 2, 3, 0, 2, 5, 3},
    { 1, 4, 0, 1, 5, 4}, { 4, 2, 5,-1,-1,-1}, { 4, 5, 2,-1,-1,-1},
    { 4, 1, 0, 4, 5, 1}, { 3, 2, 0, 3, 5, 2}, { 1, 3, 5,-1,-1,-1},
    { 4, 1, 2, 4, 3, 1}, { 3, 0, 4,-1,-1,-1}, { 2, 0, 1,-1,-1,-1},
    {-1,-1,-1,-1,-1,-1}};
__device__ const int d_num_tri[16] = {0,1,1,2,1,2,2,1,1,2,2,1,2,1,1,0};
__device__ const int d_edge_v0[6]  = {0,0,0,1,1,2};
__device__ const int d_edge_v1[6]  = {1,2,3,2,3,3};

// ---- CDNA5 async global->LDS staging helpers -------------------------------
// Flat address of a __shared__ object: low 32 bits are the LDS byte offset
// (ISA 10.2 aperture rules), which is what the async instruction's VDST
// VGPR expects.
__device__ __forceinline__ void async_load_lds_b128(void* lds_ptr, const void* gptr) {
    unsigned lds_off = (unsigned)(unsigned long long)lds_ptr;
    unsigned long long ga = (unsigned long long)gptr;
    asm volatile("global_load_async_to_lds_b128 %0, %1, off"
                 :: "v"(lds_off), "v"(ga) : "memory");
}
__device__ __forceinline__ void async_load_lds_b64(void* lds_ptr, const void* gptr) {
    unsigned lds_off = (unsigned)(unsigned long long)lds_ptr;
    unsigned long long ga = (unsigned long long)gptr;
    asm volatile("global_load_async_to_lds_b64 %0, %1, off"
                 :: "v"(lds_off), "v"(ga) : "memory");
}
__device__ __forceinline__ void wait_asynccnt0() {
    asm volatile("s_wait_asynccnt 0x0" ::: "memory");
}

// stage n u64 keys (n even except possibly a 1-element tail) into LDS tile
__device__ __forceinline__ void stage_tile(unsigned long long* tile,
                                           const unsigned long long* gsrc,
                                           int n, int tid) {
    int pairs = n >> 1;
    for (int i = tid; i < pairs; i += RS_THREADS)
        async_load_lds_b128(&tile[2 * i], gsrc + 2 * i);
    if ((n & 1) && tid == 0)
        async_load_lds_b64(&tile[n - 1], gsrc + (n - 1));
    wait_asynccnt0();
}

// ===========================================================================
// 1) pos = verts_grid + tanh(deform)/GRID_RES ; occ = sdf > 0
// ===========================================================================
__global__ void k_pos_occ(const float* __restrict__ vg,
                          const float* __restrict__ sdf,
                          const float* __restrict__ def,
                          float* __restrict__ pos,
                          unsigned char* __restrict__ occ) {
    int i = blockIdx.x * blockDim.x + threadIdx.x;
    if (i < 3 * NVERTS)
        pos[i] = vg[i] + (1.0f / GRID_RES) * tanhf(def[i]);
    if (i < NVERTS)
        occ[i] = (sdf[i] > 0.0f) ? 1 : 0;
}

// ===========================================================================
// 2) packed edge keys: (key<<24)|slot, key = a*NVERTS+b or SENT
// ===========================================================================
__global__ void k_build_keys(const int* __restrict__ tet,
                             const unsigned char* __restrict__ occ,
                             unsigned long long* __restrict__ keys) {
    int t = blockIdx.x * blockDim.x + threadIdx.x;
    if (t >= NTETS) return;
    if (t + 4096 < NTETS) __builtin_prefetch(tet + 4 * (t + 4096), 0, 0);

    int v[4]; int o[4]; int s = 0;
#pragma unroll
    for (int j = 0; j < 4; ++j) { v[j] = tet[t * 4 + j]; o[j] = occ[v[j]]; s += o[j]; }
    bool valid = (s > 0) && (s < 4);
#pragma unroll
    for (int e = 0; e < 6; ++e) {
        int va = v[d_edge_v0[e]], vb = v[d_edge_v1[e]];
        int a = va < vb ? va : vb;
        int b = va < vb ? vb : va;
        unsigned long long key =
            valid ? ((unsigned long long)a * NVERTS + (unsigned long long)b) : SENT_KEY;
        keys[t * 6 + e] = (key << 24) | (unsigned)(t * 6 + e);
    }
}

// ===========================================================================
// 3a) radix histogram (async-staged tile in LDS)
// ===========================================================================
__global__ __launch_bounds__(RS_THREADS)
void k_radix_hist(const unsigned long long* __restrict__ keys,
                  unsigned int* __restrict__ hist, int shift) {
    __shared__ __align__(16) unsigned long long tile[RS_TILE];   // 32 KB
    __shared__ unsigned int lhist[RS_DIGITS];                    // 1 KB
    int tid  = threadIdx.x;
    int base = blockIdx.x * RS_TILE;
    int n    = KEDGE - base; if (n > RS_TILE) n = RS_TILE;

    for (int d = tid; d < RS_DIGITS; d += RS_THREADS) lhist[d] = 0;

    stage_tile(tile, keys + base, n, tid);
    __syncthreads();

    for (int i = tid; i < n; i += RS_THREADS) {
        unsigned d = (unsigned)((tile[i] >> shift) & 0xFF);
        atomicAdd(&lhist[d], 1u);
    }
    __syncthreads();
    for (int d = tid; d < RS_DIGITS; d += RS_THREADS)
        hist[d * RS_BLOCKS + blockIdx.x] = lhist[d];
}

// ===========================================================================
// 3b) hierarchical exclusive scan: local -> tiny single-block -> add
// ===========================================================================
__global__ __launch_bounds__(256)
void k_scan_local(unsigned int* __restrict__ data,
                  unsigned int* __restrict__ partials, int total) {
    __shared__ unsigned int sbuf[256];
    int tid   = threadIdx.x;
    int start = blockIdx.x * SCAN_TILE + tid * 16;

    unsigned vals[16];
    unsigned sum = 0;
#pragma unroll
    for (int j = 0; j < 16; ++j) {
        int i = start + j;
        vals[j] = (i < total) ? data[i] : 0u;
        sum += vals[j];
    }
    sbuf[tid] = sum;
    __syncthreads();
    for (int off = 1; off < 256; off <<= 1) {
        unsigned add = (tid >= off) ? sbuf[tid - off] : 0u;
        __syncthreads();
        sbuf[tid] += add;
        __syncthreads();
    }
    unsigned run = sbuf[tid] - sum;   // block-local exclusive base for this thread
#pragma unroll
    for (int j = 0; j < 16; ++j) {
        int i = start + j;
        if (i < total) data[i] = run;
        run += vals[j];
    }
    if (tid == 255) partials[blockIdx.x] = sbuf[255];   // block total
}

__global__ __launch_bounds__(256)
void k_scan_exclusive(unsigned int* __restrict__ data, int total) {
    __shared__ unsigned int buf[256];
    __shared__ unsigned int carry;
    int tid = threadIdx.x;
    if (tid == 0) carry = 0;
    __syncthreads();
    for (int base = 0; base < total; base += 256) {
        int i = base + tid;
        unsigned v = (i < total) ? data[i] : 0u;
        buf[tid] = v;
        __syncthreads();
        for (int off = 1; off < 256; off <<= 1) {
            unsigned add = (tid >= off) ? buf[tid - off] : 0u;
            __syncthreads();
            buf[tid] += add;
            __syncthreads();
        }
        unsigned incl = buf[tid];
        unsigned c    = carry;
        if (i < total) data[i] = c + incl - v;
        __syncthreads();
        if (tid == 255) carry = c + incl;
        __syncthreads();
    }
}

__global__ void k_scan_add(unsigned int* __restrict__ data,
                           const unsigned int* __restrict__ partials, int total) {
    int i = blockIdx.x * blockDim.x + threadIdx.x;
    if (i < total) data[i] += partials[i / SCAN_TILE];
}

// ===========================================================================
// 3c) stable radix scatter: wave32 ballot multisplit, async-staged tile
// ===========================================================================
__global__ __launch_bounds__(RS_THREADS)
void k_radix_scatter(const unsigned long long* __restrict__ keys,
                     unsigned long long* __restrict__ out,
                     const unsigned int* __restrict__ hist, int shift) {
    __shared__ __align__(16) unsigned long long tile[RS_TILE];   // 32 KB
    __shared__ unsigned int digitBase[RS_DIGITS];                // running global base
    __shared__ unsigned int waveCnt[8][RS_DIGITS];               // 8 KB
    __shared__ unsigned int roundTotal[RS_DIGITS];               // 1 KB
    int tid  = threadIdx.x;
    int wave = tid >> 5;
    int lane = tid & 31;
    int base = blockIdx.x * RS_TILE;
    int n    = KEDGE - base; if (n > RS_TILE) n = RS_TILE;

    for (int d = tid; d < RS_DIGITS; d += RS_THREADS)
        digitBase[d] = hist[d * RS_BLOCKS + blockIdx.x];

    stage_tile(tile, keys + base, n, tid);
    __syncthreads();

    for (int round = 0; round < RS_ITEMS; ++round) {
        int  i      = round * RS_THREADS + tid;    // block-local order = memory order
        bool active = (i < n);
        unsigned long long kv = active ? tile[i] : 0ull;
        unsigned dig = (unsigned)((kv >> shift) & 0xFF);

        {
#pragma unroll
            for (int w = 0; w < 8; ++w) waveCnt[w][tid] = 0;
        }
        __syncthreads();

        // lanes in this wave holding the same digit
        unsigned peers = 0xFFFFFFFFu;
#pragma unroll
        for (int b = 0; b < 8; ++b) {
            bool p = (dig >> b) & 1u;
            unsigned bal = (unsigned)__ballot(p);
            peers &= p ? bal : ~bal;
        }
        peers &= (unsigned)__ballot(active);
        unsigned ltmask     = (lane == 0) ? 0u : (0xFFFFFFFFu >> (32 - lane));
        unsigned rankInWave = __popc(peers & ltmask);
        if (active && rankInWave == 0)
            waveCnt[wave][dig] = __popc(peers);
        __syncthreads();

        // thread d: exclusive wave prefix + round total for digit d
        {
            unsigned run = 0;
#pragma unroll
            for (int w = 0; w < 8; ++w) {
                unsigned c = waveCnt[w][tid];
                waveCnt[w][tid] = run;
                run += c;
            }
            roundTotal[tid] = run;
        }
        __syncthreads();

        if (active) {
            unsigned p = digitBase[dig] + waveCnt[wave][dig] + rankInWave;
            out[p] = kv;
        }
        __syncthreads();
        digitBase[tid] += roundTotal[tid];
        __syncthreads();
    }
}

// ===========================================================================
// 4) unique-rank pipeline: fill uq with SENT, per-block head counts,
//    block-sum scan, then emit uq/inv.
// ===========================================================================
__global__ void k_fill_uq(unsigned long long* __restrict__ uq) {
    int u = blockIdx.x * blockDim.x + threadIdx.x;
    if (u < KEDGE) uq[u] = SENT_KEY;
}

__device__ __forceinline__ bool is_head(const unsigned long long* s, int i) {
    if (i == 0) return true;
    return (s[i] >> 24) != (s[i - 1] >> 24);
}

__global__ __launch_bounds__(256)
void k_head_count(const unsigned long long* __restrict__ sorted,
                  unsigned int* __restrict__ partials) {
    __shared__ unsigned int tsum[256];
    int tid   = threadIdx.x;
    int start = blockIdx.x * RS_TILE + tid * RS_ITEMS;
    unsigned cnt = 0;
    for (int j = 0; j < RS_ITEMS; ++j) {
        int i = start + j;
        if (i < KEDGE && is_head(sorted, i)) ++cnt;
    }
    tsum[tid] = cnt;
    __syncthreads();
    for (int off = 128; off > 0; off >>= 1) {
        if (tid < off) tsum[tid] += tsum[tid + off];
        __syncthreads();
    }
    if (tid == 0) partials[blockIdx.x] = tsum[0];
}

__global__ __launch_bounds__(256)
void k_emit_unique(const unsigned long long* __restrict__ sorted,
                   const unsigned int* __restrict__ partials,
                   unsigned int* __restrict__ inv,
                   unsigned long long* __restrict__ uq) {
    __shared__ unsigned int sbuf[256];
    int tid   = threadIdx.x;
    int start = blockIdx.x * RS_TILE + tid * RS_ITEMS;

    unsigned cnt = 0;
    for (int j = 0; j < RS_ITEMS; ++j) {
        int i = start + j;
        if (i < KEDGE && is_head(sorted, i)) ++cnt;
    }
    sbuf[tid] = cnt;
    __syncthreads();
    for (int off = 1; off < 256; off <<= 1) {
        unsigned add = (tid >= off) ? sbuf[tid - off] : 0u;
        __syncthreads();
        sbuf[tid] += add;
        __syncthreads();
    }
    unsigned excl = sbuf[tid] - cnt;
    unsigned r    = partials[blockIdx.x] + excl;

    for (int j = 0; j < RS_ITEMS; ++j) {
        int i = start + j;
        if (i >= KEDGE) break;
        unsigned long long kv  = sorted[i];
        unsigned long long key = kv >> 24;
        unsigned rank;
        if (is_head(sorted, i)) { rank = r; uq[rank] = key; ++r; }
        else                    { rank = r - 1; }
        inv[(unsigned)(kv & 0xFFFFFFu)] = rank;
    }
}

// ===========================================================================
// 5) interpolated verts + per-unique crossing mask
// ===========================================================================
__global__ void k_verts(const unsigned long long* __restrict__ uq,
                        const float* __restrict__ pos,
                        const float* __restrict__ sdf,
                        const unsigned char* __restrict__ occ,
                        float* __restrict__ verts_out,
                        unsigned char* __restrict__ mask_arr) {
    int u = blockIdx.x * blockDim.x + threadIdx.x;
    if (u >= KEDGE) return;
    unsigned long long key = uq[u];
    if (key >= SENT_KEY) {
        verts_out[u * 3 + 0] = 0.0f;
        verts_out[u * 3 + 1] = 0.0f;
        verts_out[u * 3 + 2] = 0.0f;
        mask_arr[u] = 0;
        return;
    }
    int ua = (int)(key / NVERTS);
    int ub = (int)(key % NVERTS);
    bool m = ((occ[ua] ^ occ[ub]) != 0);
    float sa = sdf[ua], sb = sdf[ub];
    float denom = m ? (sa - sb) : 1.0f;
    float wa = -sb / denom;
    float wb =  sa / denom;
    mask_arr[u] = m ? 1 : 0;
#pragma unroll
    for (int c = 0; c < 3; ++c) {
        float v = pos[ua * 3 + c] * wa + pos[ub * 3 + c] * wb;
        verts_out[u * 3 + c] = m ? v : 0.0f;
    }
}

// ===========================================================================
// 6) faces (emitted as float; unique ids < 2^24 are exact in f32)
// ===========================================================================
__global__ void k_faces(const int* __restrict__ tet,
                        const unsigned char* __restrict__ occ,
                        const unsigned int* __restrict__ inv,
                        const unsigned char* __restrict__ mask_arr,
                        float* __restrict__ faces_out) {
    int t = blockIdx.x * blockDim.x + threadIdx.x;
    if (t >= NTETS) return;
    if (t + 4096 < NTETS) __builtin_prefetch(tet + 4 * (t + 4096), 0, 0);

    int o[4]; int s = 0;
#pragma unroll
    for (int j = 0; j < 4; ++j) { o[j] = occ[tet[t * 4 + j]]; s += o[j]; }
    bool valid   = (s > 0) && (s < 4);
    int tetindex = o[0] + 2 * o[1] + 4 * o[2] + 8 * o[3];
    int ntri     = valid ? d_num_tri[tetindex] : 0;

#pragma unroll
    for (int j = 0; j < 6; ++j) {
        int trij = d_tri_table[tetindex][j];
        int col  = trij < 0 ? 0 : (trij > 5 ? 5 : trij);
        unsigned e = inv[t * 6 + col];
        int g = mask_arr[e] ? (int)e : -1;
        int k = j / 3;
        bool ok = (k < ntri) && (trij >= 0);
        faces_out[t * 6 + j] = (float)(ok ? g : -1);
    }
}

// ===========================================================================
// launcher
// ===========================================================================
extern "C" void kernel_launch(void* const* d_in, const int* in_sizes, int n_in,
                              void* d_out, int out_size, void* d_ws, size_t ws_size,
                              hipStream_t stream) {
    (void)in_sizes; (void)n_in; (void)out_size; (void)ws_size;

    const float* verts_grid = (const float*)d_in[0];
    const float* sdf        = (const float*)d_in[1];
    const float* deform     = (const float*)d_in[2];
    const int*   tet        = (const int*)d_in[3];

    // workspace carve-out (requires ~268 MB)
    char* ws = (char*)d_ws;
    size_t cur = 0;
    auto take = [&](size_t bytes) -> void* {
        void* p = ws + cur;
        cur += (bytes + 255) & ~(size_t)255;
        return p;
    };
    float*              pos      = (float*)take((size_t)3 * NVERTS * 4);
    unsigned char*      occ      = (unsigned char*)take((size_t)NVERTS);
    unsigned long long* keysA    = (unsigned long long*)take((size_t)KEDGE * 8);
    unsigned long long* keysB    = (unsigned long long*)take((size_t)KEDGE * 8);
    unsigned int*       hist     = (unsigned int*)take((size_t)RS_DIGITS * RS_BLOCKS * 4);
    unsigned int*       partials = (unsigned int*)take((size_t)RS_BLOCKS * 4);
    unsigned int*       inv      = (unsigned int*)take((size_t)KEDGE * 4);
    unsigned long long* uq       = (unsigned long long*)take((size_t)KEDGE * 8);
    unsigned char*      mask_arr = (unsigned char*)take((size_t)KEDGE);

    float* verts_out = (float*)d_out;                       // K x 3
    float* faces_out = (float*)d_out + (size_t)KEDGE * 3;   // NTETS x 2 x 3

    const int TPB      = 256;
    const int gridPos  = (3 * NVERTS + TPB - 1) / TPB;
    const int gridTets = (NTETS + TPB - 1) / TPB;
    const int gridK    = (KEDGE + TPB - 1) / TPB;

    const int histTotal  = RS_DIGITS * RS_BLOCKS;                       // 562,688
    const int histBlocks = (histTotal + SCAN_TILE - 1) / SCAN_TILE;     // 138
    const int histAddGrid = (histTotal + TPB - 1) / TPB;

    k_pos_occ<<<gridPos, TPB, 0, stream>>>(verts_grid, sdf, deform, pos, occ);
    k_build_keys<<<gridTets, TPB, 0, stream>>>(tet, occ, keysA);

    unsigned long long* src = keysA;
    unsigned long long* dst = keysB;
    for (int p = 0; p < RS_PASSES; ++p) {
        int shift = RS_SHIFT0 + 8 * p;
        k_radix_hist<<<RS_BLOCKS, RS_THREADS, 0, stream>>>(src, hist, shift);
        // hierarchical exclusive scan of digit-major histogram
        k_scan_local<<<histBlocks, 256, 0, stream>>>(hist, partials, histTotal);
        k_scan_exclusive<<<1, 256, 0, stream>>>(partials, histBlocks);
        k_scan_add<<<histAddGrid, TPB, 0, stream>>>(hist, partials, histTotal);
        k_radix_scatter<<<RS_BLOCKS, RS_THREADS, 0, stream>>>(src, dst, hist, shift);
        unsigned long long* tmp = src; src = dst; dst = tmp;
    }
    unsigned long long* sorted = src;   // RS_PASSES is odd -> keysB

    k_fill_uq<<<gridK, TPB, 0, stream>>>(uq);
    k_head_count<<<RS_BLOCKS, RS_THREADS, 0, stream>>>(sorted, partials);
    k_scan_exclusive<<<1, 256, 0, stream>>>(partials, RS_BLOCKS);
    k_emit_unique<<<RS_BLOCKS, RS_THREADS, 0, stream>>>(sorted, partials, inv, uq);

    k_verts<<<gridK, TPB, 0, stream>>>(uq, pos, sdf, occ, verts_out, mask_arr);
    k_faces<<<gridTets, TPB, 0, stream>>>(tet, occ, inv, mask_arr, faces_out);
}